// AttentionDecoder_52329881534731
// MI455X (gfx1250) — compile-verified
//
#include <hip/hip_runtime.h>
#include <hip/hip_bf16.h>
#include <math.h>

typedef __attribute__((ext_vector_type(2))) float v2f;
typedef __attribute__((ext_vector_type(8))) float v8f;
typedef __attribute__((ext_vector_type(4))) int   v4i;
typedef __attribute__((address_space(1))) v4i gv4i;   // global int4
typedef __attribute__((address_space(3))) v4i lv4i;   // LDS int4

#define Bq   64
#define Pq   196
#define ENCq 2048
#define Lq   20
#define Vq   32000
#define Hq   512
#define Eq   512
#define Aq   512
#define Tq   (Lq - 1)

#define KCHUNK 256
#define KPAD   4
#define BLOCK_N 256   // 8 waves x 32 cols (2 x 16x16 tiles per wave)

#if __has_builtin(__builtin_amdgcn_global_load_async_to_lds_b128) && \
    __has_builtin(__builtin_amdgcn_s_wait_asynccnt)
#define USE_ASYNC_LDS 1
#else
#define USE_ASYNC_LDS 0
#endif

// ---------------------------------------------------------------------------
// Generic tiled fp32 WMMA GEMM:  Y[M,N] = (accumulate? Y : 0) + X[M,K] @ W[N,K]^T + bias
// Optional row_map: X source-row indirection.  Optional declen mask: rows with
// dec_len[row] <= tstep get 0 written (masked predictions).
// One block = 16 rows x 256 cols; each of 8 waves owns a 16x32 tile (two
// independent V_WMMA_F32_16X16X4_F32 accumulation chains for ILP).
// A panel double-buffered in LDS via async global->LDS DMA (ASYNCcnt),
// B streamed from global as float2 pairs matching the f32 WMMA lane layout.
// Requires K % KCHUNK == 0, M % 16 == 0 (true for all launches here).
// ---------------------------------------------------------------------------
__global__ void __launch_bounds__(256)
wmma_gemm_xwT(const float* __restrict__ X, int ldx,
              const int* __restrict__ row_map,
              const float* __restrict__ W, int ldw,
              const float* __restrict__ bias,
              float* __restrict__ Y, long long ldy,
              int M, int N, int K,
              const int* __restrict__ declen, int tstep,
              int accumulate)
{
    __shared__ __align__(16) float As[2][16][KCHUNK + KPAD];

    const int tid   = threadIdx.x;
    const int wave  = tid >> 5;
    const int lane  = tid & 31;
    const int mbase = blockIdx.y * 16;
    const int nt0   = blockIdx.x * BLOCK_N + wave * 32;    // wave's first N tile
    const bool wave_active = (nt0 < N);                    // N % 256 == 0 here

    const int frag_rc = lane & 15;            // A row / B col for this lane
    const int koff    = (lane >> 4) << 1;     // 0 (lanes 0-15) or 2 (lanes 16-31)

    v8f acc0 = {};                            // two 16x16 f32 accumulators
    v8f acc1 = {};

    const float* W0 = W + (size_t)(nt0 + frag_rc) * (size_t)ldw;
    const float* W1 = W0 + (size_t)16 * (size_t)ldw;

    // per-thread A staging slots: 4 float4 each (16 rows x 256 k per chunk)
    const float* srcp[4];
    unsigned     loff[4];
    #pragma unroll
    for (int i = 0; i < 4; ++i) {
        int q  = tid + i * 256;               // float4 index within chunk
        int r  = q >> 6;                      // 64 float4 per row
        int kf = (q & 63) << 2;
        int gm = mbase + r;
        int src = row_map ? row_map[gm] : gm;
        srcp[i] = X + (size_t)src * (size_t)ldx + kf;
        loff[i] = (unsigned)((r * (KCHUNK + KPAD) + kf) * 4);   // bytes in buffer
    }

    const int nchunks = K / KCHUNK;

#if USE_ASYNC_LDS
    {   // prologue: async DMA chunk 0 into buffer 0
        char* lb = (char*)&As[0][0][0];
        #pragma unroll
        for (int i = 0; i < 4; ++i)
            __builtin_amdgcn_global_load_async_to_lds_b128(
                (gv4i*)(void*)srcp[i], (lv4i*)(void*)(lb + loff[i]), 0, 0);
    }
#endif

    for (int ck = 0; ck < nchunks; ++ck) {
        const int buf = ck & 1;
        const int kb  = ck * KCHUNK;
#if USE_ASYNC_LDS
        if (ck + 1 < nchunks) {
            // stream next chunk into the other buffer, then wait for the 4
            // older (in-order) DMAs of the current chunk to complete.
            char* lb = (char*)&As[buf ^ 1][0][0];
            const int kb2 = kb + KCHUNK;
            #pragma unroll
            for (int i = 0; i < 4; ++i)
                __builtin_amdgcn_global_load_async_to_lds_b128(
                    (gv4i*)(void*)(srcp[i] + kb2), (lv4i*)(void*)(lb + loff[i]), 0, 0);
            __builtin_amdgcn_s_wait_asynccnt(4);
        } else {
            __builtin_amdgcn_s_wait_asynccnt(0);
        }
#else
        {   // synchronous fallback staging
            char* lb = (char*)&As[buf][0][0];
            #pragma unroll
            for (int i = 0; i < 4; ++i) {
                const float4 v = *(const float4*)(srcp[i] + kb);
                float* d = (float*)(lb + loff[i]);
                d[0] = v.x; d[1] = v.y; d[2] = v.z; d[3] = v.w;
            }
        }
#endif
        __syncthreads();

        if (wave_active) {
            const float* wp0 = W0 + kb + koff;
            const float* wp1 = W1 + kb + koff;
            if (kb + KCHUNK < K)
                __builtin_prefetch(wp0 + KCHUNK, 0, 0);   // global_prefetch_b8
            #pragma unroll 4
            for (int kk = 0; kk < KCHUNK; kk += 4) {
                const float2 av = *(const float2*)(&As[buf][frag_rc][kk + koff]);
                v2f a; a.x = av.x; a.y = av.y;
                const float2 b0v = *(const float2*)(wp0 + kk);
                const float2 b1v = *(const float2*)(wp1 + kk);
                v2f b0; b0.x = b0v.x; b0.y = b0v.y;
                v2f b1; b1.x = b1v.x; b1.y = b1v.y;
                acc0 = __builtin_amdgcn_wmma_f32_16x16x4_f32(
                           false, a, false, b0, (short)0, acc0, false, false);
                acc1 = __builtin_amdgcn_wmma_f32_16x16x4_f32(
                           false, a, false, b1, (short)0, acc1, false, false);
            }
        }
        __syncthreads();
    }

    if (!wave_active) return;

    // C/D layout: VGPR r -> row r (lanes 0-15) / row r+8 (lanes 16-31), col = lane&15
    const int col   = lane & 15;
    const int rbase = (lane < 16) ? 0 : 8;
    const int n0    = nt0 + col;
    const int n1    = nt0 + 16 + col;
    #pragma unroll
    for (int r = 0; r < 8; ++r) {
        int gm = mbase + rbase + r;
        bool zero = declen && !(declen[gm] > tstep);
        float v0 = acc0[r], v1 = acc1[r];
        if (bias) { v0 += bias[n0]; v1 += bias[n1]; }
        float* yp0 = Y + (size_t)gm * (size_t)ldy + n0;
        float* yp1 = Y + (size_t)gm * (size_t)ldy + n1;
        if (accumulate) { v0 += *yp0; v1 += *yp1; }
        if (zero) { v0 = 0.0f; v1 = 0.0f; }
        *yp0 = v0;
        *yp1 = v1;
    }
}

// ---------------------------------------------------------------------------
// Sort batch by caption length (descending, stable). One block, 64 threads.
// Also emits dec_len and sort_idx (as float) into the output tail.
// ---------------------------------------------------------------------------
__global__ void sort_kernel(const int* __restrict__ y,
                            int* __restrict__ sort_idx, int* __restrict__ dec_len,
                            float* __restrict__ out_tail)
{
    __shared__ int lens[Bq];
    int b = threadIdx.x;
    if (b < Bq) {
        int c = 0;
        for (int l = 0; l < Lq; ++l) c += (y[b * Lq + l] != 0) ? 1 : 0;
        lens[b] = c;
    }
    __syncthreads();
    if (b < Bq) {
        int myl = lens[b];
        int rank = 0;
        for (int j = 0; j < Bq; ++j) {
            int lj = lens[j];
            if (lj > myl || (lj == myl && j < b)) rank++;
        }
        sort_idx[rank] = b;
        dec_len[rank]  = myl - 1;
        out_tail[rank]      = (float)(myl - 1);  // dec_len (float)
        out_tail[Bq + rank] = (float)b;          // sort_idx (float)
    }
}

__global__ void rowmap_kernel(const int* __restrict__ sort_idx, int* __restrict__ row_map)
{
    int r = blockIdx.x * 256 + threadIdx.x;
    if (r < Bq * Pq) row_map[r] = sort_idx[r / Pq] * Pq + (r % Pq);
}

__global__ void mean_kernel(const float* __restrict__ enc,
                            const int* __restrict__ sort_idx,
                            float* __restrict__ mean_enc)
{
    int b = blockIdx.x;
    int src = sort_idx[b];
    const float* base = enc + (size_t)src * Pq * ENCq;
    for (int e = threadIdx.x; e < ENCq; e += 256) {
        float s = 0.f;
        for (int p = 0; p < Pq; ++p) s += base[(size_t)p * ENCq + e];
        mean_enc[(size_t)b * ENCq + e] = s * (1.0f / (float)Pq);
    }
}

// ---------------------------------------------------------------------------
// Fused attention: score = relu(att1 + att2) . attW + attb ; softmax over P ;
// context = alpha @ enc ; masked alphas to output. One block per batch row.
// ---------------------------------------------------------------------------
__global__ void __launch_bounds__(256)
attn_kernel(const float* __restrict__ att1, const float* __restrict__ att2,
            const float* __restrict__ attW, const float* __restrict__ attb,
            const float* __restrict__ enc, const int* __restrict__ sort_idx,
            const int* __restrict__ dec_len, int t,
            float* __restrict__ context, float* __restrict__ out_alphas)
{
    __shared__ float sScore[Pq];
    __shared__ float sRed[8];
    int b    = blockIdx.x;
    int tid  = threadIdx.x;
    int wave = tid >> 5, lane = tid & 31;

    const float* a1 = att1 + (size_t)b * Pq * Aq;
    const float* a2 = att2 + (size_t)b * Aq;

    for (int p = wave; p < Pq; p += 8) {
        const float* a1p = a1 + (size_t)p * Aq;
        float s = 0.f;
        for (int a = lane; a < Aq; a += 32) {
            float v = a1p[a] + a2[a];
            v = v > 0.f ? v : 0.f;
            s += v * attW[a];
        }
        for (int off = 16; off > 0; off >>= 1) s += __shfl_xor(s, off, 32);
        if (lane == 0) sScore[p] = s + attb[0];
    }
    __syncthreads();

    // softmax over P
    float lm = -1e30f;
    for (int p = tid; p < Pq; p += 256) lm = fmaxf(lm, sScore[p]);
    for (int off = 16; off > 0; off >>= 1) lm = fmaxf(lm, __shfl_xor(lm, off, 32));
    if (lane == 0) sRed[wave] = lm;
    __syncthreads();
    float m = sRed[0];
    for (int w = 1; w < 8; ++w) m = fmaxf(m, sRed[w]);

    float ls = 0.f;
    for (int p = tid; p < Pq; p += 256) ls += __expf(sScore[p] - m);
    for (int off = 16; off > 0; off >>= 1) ls += __shfl_xor(ls, off, 32);
    __syncthreads();
    if (lane == 0) sRed[wave] = ls;
    __syncthreads();
    float sum = 0.f;
    for (int w = 0; w < 8; ++w) sum += sRed[w];
    float inv = 1.0f / sum;
    __syncthreads();

    for (int p = tid; p < Pq; p += 256) sScore[p] = __expf(sScore[p] - m) * inv;
    __syncthreads();

    bool active = dec_len[b] > t;
    for (int p = tid; p < Pq; p += 256)
        out_alphas[(size_t)b * Tq * Pq + (size_t)t * Pq + p] = active ? sScore[p] : 0.0f;

    int src = sort_idx[b];
    const float* eb = enc + (size_t)src * Pq * ENCq;
    for (int e = tid; e < ENCq; e += 256) {
        float s = 0.f;
        for (int p = 0; p < Pq; ++p) s += sScore[p] * eb[(size_t)p * ENCq + e];
        context[(size_t)b * ENCq + e] = s;
    }
}

// x_in = [ embed(tok_t) , sigmoid(gateY) * context ]
__global__ void xin_kernel(const float* __restrict__ embed_table,
                           const int* __restrict__ y, const int* __restrict__ sort_idx,
                           const float* __restrict__ gateY, const float* __restrict__ context,
                           int t, float* __restrict__ x_in)
{
    int b = blockIdx.x;
    int tok = y[sort_idx[b] * Lq + t];
    const float* emb = embed_table + (size_t)tok * Eq;
    float* xb = x_in + (size_t)b * (Eq + ENCq);
    for (int e = threadIdx.x; e < Eq; e += 256) xb[e] = emb[e];
    const float* gb = gateY  + (size_t)b * ENCq;
    const float* cb = context + (size_t)b * ENCq;
    for (int e = threadIdx.x; e < ENCq; e += 256)
        xb[Eq + e] = (1.0f / (1.0f + __expf(-gb[e]))) * cb[e];
}

// elementwise LSTM cell (torch gate order i,f,g,o), masked state update
__global__ void lstm_kernel(const float* __restrict__ gates,
                            const int* __restrict__ dec_len, int t,
                            float* __restrict__ h, float* __restrict__ c,
                            float* __restrict__ h_new)
{
    int idx = blockIdx.x * 256 + threadIdx.x;
    if (idx >= Bq * Hq) return;
    int b = idx / Hq, j = idx % Hq;
    const float* g = gates + (size_t)b * 4 * Hq;
    float ig = g[j], fg = g[Hq + j], gg = g[2 * Hq + j], og = g[3 * Hq + j];
    float si = 1.f / (1.f + __expf(-ig));
    float sf = 1.f / (1.f + __expf(-fg));
    float so = 1.f / (1.f + __expf(-og));
    float cn = sf * c[idx] + si * tanhf(gg);
    float hn = so * tanhf(cn);
    h_new[idx] = hn;
    if (dec_len[b] > t) { h[idx] = hn; c[idx] = cn; }
}

// ---------------------------------------------------------------------------
extern "C" void kernel_launch(void* const* d_in, const int* in_sizes, int n_in,
                              void* d_out, int out_size, void* d_ws, size_t ws_size,
                              hipStream_t stream)
{
    (void)in_sizes; (void)n_in; (void)out_size; (void)ws_size;
    const float* enc          = (const float*)d_in[0];
    const int*   y            = (const int*)  d_in[1];
    const float* embed_table  = (const float*)d_in[2];
    const float* enc2att_W    = (const float*)d_in[3];
    const float* hidden2att_W = (const float*)d_in[4];
    const float* att_W        = (const float*)d_in[5];
    const float* att_b        = (const float*)d_in[6];
    const float* f_beta_W     = (const float*)d_in[7];
    const float* f_beta_b     = (const float*)d_in[8];
    const float* W_ih         = (const float*)d_in[9];
    const float* b_ih         = (const float*)d_in[10];
    const float* W_hh         = (const float*)d_in[11];
    const float* b_hh         = (const float*)d_in[12];
    const float* fc_W         = (const float*)d_in[13];
    const float* fc_b         = (const float*)d_in[14];
    const float* init_h_W     = (const float*)d_in[15];
    const float* init_h_b     = (const float*)d_in[16];
    const float* init_c_W     = (const float*)d_in[17];
    const float* init_c_b     = (const float*)d_in[18];

    float* out = (float*)d_out;
    // output layout: preds [B,T,V] | alphas [B,T,P] | dec_len [B] | sort_idx [B]
    const size_t OUT_ALPHA = (size_t)Bq * Tq * Vq;                 // 38,912,000
    const size_t OUT_TAIL  = OUT_ALPHA + (size_t)Bq * Tq * Pq;     // + 238,336

    // workspace layout (floats)
    float* ws = (float*)d_ws;
    float* att1   = ws;                                   // 12544*512
    float* att2   = att1 + (size_t)Bq * Pq * Aq;          // 64*512
    float* ctx    = att2 + (size_t)Bq * Aq;               // 64*2048
    float* gateY  = ctx  + (size_t)Bq * ENCq;             // 64*2048
    float* x_in   = gateY+ (size_t)Bq * ENCq;             // 64*2560
    float* gates  = x_in + (size_t)Bq * (Eq + ENCq);      // 64*2048
    float* h      = gates+ (size_t)Bq * 4 * Hq;           // 64*512
    float* c      = h    + (size_t)Bq * Hq;               // 64*512
    float* h_new  = c    + (size_t)Bq * Hq;               // 64*512
    float* meanE  = h_new+ (size_t)Bq * Hq;               // 64*2048
    int*   iws      = (int*)(meanE + (size_t)Bq * ENCq);
    int*   sort_idx = iws;
    int*   dec_len  = iws + Bq;
    int*   row_map  = iws + 2 * Bq;

    // ---- setup ----
    sort_kernel<<<1, 64, 0, stream>>>(y, sort_idx, dec_len, out + OUT_TAIL);
    rowmap_kernel<<<(Bq * Pq + 255) / 256, 256, 0, stream>>>(sort_idx, row_map);
    mean_kernel<<<Bq, 256, 0, stream>>>(enc, sort_idx, meanE);

    // h0 / c0 : [64,2048] @ [512,2048]^T
    {
        dim3 g((Hq + BLOCK_N - 1) / BLOCK_N, Bq / 16);
        wmma_gemm_xwT<<<g, 256, 0, stream>>>(meanE, ENCq, nullptr, init_h_W, ENCq,
                                             init_h_b, h, Hq, Bq, Hq, ENCq,
                                             nullptr, 0, 0);
        wmma_gemm_xwT<<<g, 256, 0, stream>>>(meanE, ENCq, nullptr, init_c_W, ENCq,
                                             init_c_b, c, Hq, Bq, Hq, ENCq,
                                             nullptr, 0, 0);
    }
    // att1 : [12544,2048] @ [512,2048]^T  (rows gathered through sort order)
    {
        dim3 g((Aq + BLOCK_N - 1) / BLOCK_N, (Bq * Pq) / 16);
        wmma_gemm_xwT<<<g, 256, 0, stream>>>(enc, ENCq, row_map, enc2att_W, ENCq,
                                             nullptr, att1, Aq, Bq * Pq, Aq, ENCq,
                                             nullptr, 0, 0);
    }

    // ---- sequential decode ----
    for (int t = 0; t < Tq; ++t) {
        // att2 = h @ hidden2att_W^T : [64,512]x[512,512]
        {
            dim3 g((Aq + BLOCK_N - 1) / BLOCK_N, Bq / 16);
            wmma_gemm_xwT<<<g, 256, 0, stream>>>(h, Hq, nullptr, hidden2att_W, Hq,
                                                 nullptr, att2, Aq, Bq, Aq, Hq,
                                                 nullptr, 0, 0);
        }
        attn_kernel<<<Bq, 256, 0, stream>>>(att1, att2, att_W, att_b, enc, sort_idx,
                                            dec_len, t, ctx, out + OUT_ALPHA);
        // gateY = h @ f_beta_W^T : [64,512]x[2048,512]
        {
            dim3 g((ENCq + BLOCK_N - 1) / BLOCK_N, Bq / 16);
            wmma_gemm_xwT<<<g, 256, 0, stream>>>(h, Hq, nullptr, f_beta_W, Hq,
                                                 f_beta_b, gateY, ENCq, Bq, ENCq, Hq,
                                                 nullptr, 0, 0);
        }
        xin_kernel<<<Bq, 256, 0, stream>>>(embed_table, y, sort_idx, gateY, ctx, t, x_in);
        // gates = x_in @ W_ih^T + b_ih  ;  += h @ W_hh^T + b_hh
        {
            dim3 g((4 * Hq + BLOCK_N - 1) / BLOCK_N, Bq / 16);
            wmma_gemm_xwT<<<g, 256, 0, stream>>>(x_in, Eq + ENCq, nullptr, W_ih, Eq + ENCq,
                                                 b_ih, gates, 4 * Hq, Bq, 4 * Hq, Eq + ENCq,
                                                 nullptr, 0, 0);
            wmma_gemm_xwT<<<g, 256, 0, stream>>>(h, Hq, nullptr, W_hh, Hq,
                                                 b_hh, gates, 4 * Hq, Bq, 4 * Hq, Hq,
                                                 nullptr, 0, 1);
        }
        lstm_kernel<<<(Bq * Hq + 255) / 256, 256, 0, stream>>>(gates, dec_len, t, h, c, h_new);
        // logits -> masked predictions at out[b*T*V + t*V + v]
        {
            dim3 g((Vq + BLOCK_N - 1) / BLOCK_N, Bq / 16);
            wmma_gemm_xwT<<<g, 256, 0, stream>>>(h_new, Hq, nullptr, fc_W, Hq,
                                                 fc_b, out + (size_t)t * Vq,
                                                 (long long)Tq * Vq, Bq, Vq, Hq,
                                                 dec_len, t, 0);
        }
    }
}